// SS3D_v6_34926674051368
// MI455X (gfx1250) — compile-verified
//
#include <hip/hip_runtime.h>
#include <math.h>

// SS3D selective-scan block for gfx1250 (MI455X), fp32 end-to-end.
// GEMMs: V_WMMA_F32_16X16X4_F32 with A-fragment reuse across a 64/48-wide
// N strip per wave. Scan: GLOBAL_LOAD_ASYNC_TO_LDS_B128 chunk staging.

typedef __attribute__((ext_vector_type(2))) float v2f;
typedef __attribute__((ext_vector_type(8))) float v8f;

#define B_   2
#define S_   16
#define DM_  192
#define DI_  192
#define DSN  16          // state size
#define K_   6
#define DR_  12
#define L_   (S_*S_*S_)  // 4096
#define NCH  48          // x_dbl channels padded 44 -> 48
#define CHUNK 16         // scan LDS chunk (steps): 16*48 floats = 192 lanes * 16B

// ---------------------------------------------------------------------------
// WMMA GEMM: C[M,192] = A[M,192] * W[192,192]^T   (C[m,n] = sum_k A[m,k]W[n,k])
// One wave computes a 16x64 strip (4 accumulators); the A fragment is loaded
// once per K-step and reused for 4 WMMAs -> 5 b64 loads per 4 WMMAs.
// A-frag (32-bit A 16x4): lanes0-15 -> K=k0+0/1, lanes16-31 -> K=k0+2/3.
// D-frag (32-bit C/D 16x16): VGPR r -> row r (lanes0-15) / r+8 (lanes16-31).
// ---------------------------------------------------------------------------
__global__ __launch_bounds__(32)
void gemm_xtw_kernel(const float* __restrict__ A, const float* __restrict__ W,
                     float* __restrict__ C) {
  const int tileM = blockIdx.x;
  const int strip = blockIdx.y;            // 3 strips of 64 columns
  const int lane  = threadIdx.x;
  const int half  = (lane < 16) ? 0 : 2;
  const int lm    = lane & 15;

  const float* arow = A + (size_t)(tileM * 16 + lm) * DM_;
  const float* w0   = W + (size_t)(strip * 64 + lm) * DM_;

  v8f acc0 = {}, acc1 = {}, acc2 = {}, acc3 = {};
#pragma unroll 2
  for (int k0 = 0; k0 < DM_; k0 += 4) {
    const v2f a  = *(const v2f*)(arow + k0 + half);
    const v2f b0 = *(const v2f*)(w0 + (size_t)0  * DM_ + k0 + half);
    const v2f b1 = *(const v2f*)(w0 + (size_t)16 * DM_ + k0 + half);
    const v2f b2 = *(const v2f*)(w0 + (size_t)32 * DM_ + k0 + half);
    const v2f b3 = *(const v2f*)(w0 + (size_t)48 * DM_ + k0 + half);
    acc0 = __builtin_amdgcn_wmma_f32_16x16x4_f32(false, a, false, b0, (short)0, acc0, false, false);
    acc1 = __builtin_amdgcn_wmma_f32_16x16x4_f32(false, a, false, b1, (short)0, acc1, false, false);
    acc2 = __builtin_amdgcn_wmma_f32_16x16x4_f32(false, a, false, b2, (short)0, acc2, false, false);
    acc3 = __builtin_amdgcn_wmma_f32_16x16x4_f32(false, a, false, b3, (short)0, acc3, false, false);
  }
  const int rofs = (lane < 16) ? 0 : 8;
  float* crow = C + (size_t)(tileM * 16) * DM_ + strip * 64 + lm;
#pragma unroll
  for (int r = 0; r < 8; ++r) {
    float* cr = crow + (size_t)(r + rofs) * DM_;
    cr[0]  = acc0[r];
    cr[16] = acc1[r];
    cr[32] = acc2[r];
    cr[48] = acc3[r];
  }
}

// ---------------------------------------------------------------------------
// Depthwise 3x3x3 conv (SAME) + SiLU, scattered into the 3 scan permutations:
// xsm layout (B,3,L,DI) channel-contiguous.
//   p=0: l = h*256 + w*16 + d ; p=1: l = d*256 + w*16 + h ; p=2: l = h*256 + d*16 + w
// ---------------------------------------------------------------------------
__global__ __launch_bounds__(192)
void conv_silu_perm_kernel(const float* __restrict__ xi, const float* __restrict__ cw,
                           const float* __restrict__ cb, float* __restrict__ xsm) {
  const int bl = blockIdx.x;               // b*L + l
  const int b  = bl >> 12;
  const int l  = bl & (L_ - 1);
  const int h  = l >> 8, w = (l >> 4) & 15, dd = l & 15;
  const int di = threadIdx.x;

  float acc = cb[di];
  const float* wk = cw + di * 27;
#pragma unroll
  for (int i = 0; i < 3; ++i) {
    const int nh = h + i - 1;
    if (nh < 0 || nh >= S_) continue;
#pragma unroll
    for (int j = 0; j < 3; ++j) {
      const int nw = w + j - 1;
      if (nw < 0 || nw >= S_) continue;
#pragma unroll
      for (int kk = 0; kk < 3; ++kk) {
        const int nd = dd + kk - 1;
        if (nd < 0 || nd >= S_) continue;
        acc += wk[i * 9 + j * 3 + kk] *
               xi[((size_t)b * L_ + (nh << 8) + (nw << 4) + nd) * DI_ + di];
      }
    }
  }
  const float v = acc / (1.f + expf(-acc));   // SiLU

  const int l1 = (dd << 8) + (w << 4) + h;
  const int l2 = (h << 8) + (dd << 4) + w;
  xsm[((size_t)(b * 3 + 0) * L_ + l) * DI_ + di] = v;
  xsm[((size_t)(b * 3 + 1) * L_ + l1) * DI_ + di] = v;
  xsm[((size_t)(b * 3 + 2) * L_ + l2) * DI_ + di] = v;
}

// ---------------------------------------------------------------------------
// x_dbl (WMMA): for each (b,k): out[l, c] = sum_d xs_k[l, d] * xpw[k, c, d]
// One wave computes the full 16x48 strip (3 accumulators, A-frag reused).
// k>=3 are flipped traversals of xsm[k-3]; N padded 44->48 (cols 44..47 junk).
// Output layout (B,6,L,48): [0..11]=dt rank, [12..27]=B, [28..43]=C.
// ---------------------------------------------------------------------------
__global__ __launch_bounds__(32)
void gemm_xdbl_kernel(const float* __restrict__ xsm, const float* __restrict__ xpw,
                      float* __restrict__ xdbl) {
  const int tileM = blockIdx.x;            // 256 tiles of L
  const int bk    = blockIdx.y;            // b*6 + k
  const int b = bk / 6, k = bk % 6;
  const int p = (k >= 3) ? (k - 3) : k;
  const int lane = threadIdx.x;
  const int half = (lane < 16) ? 0 : 2;
  const int lm   = lane & 15;

  const int m    = tileM * 16 + lm;                  // scan index
  const int lrow = (k < 3) ? m : (L_ - 1 - m);       // source row (flip)
  const float* arow  = xsm + ((size_t)(b * 3 + p) * L_ + lrow) * DI_;
  const float* wbase = xpw + (size_t)k * 44 * DI_;

  const float* wrow0 = wbase + (size_t)(lm) * DI_;                      // c = 0..15
  const float* wrow1 = wbase + (size_t)(16 + lm) * DI_;                 // c = 16..31
  const int   c2     = 32 + lm;                                         // c = 32..47
  const float* wrow2 = wbase + (size_t)(c2 < 44 ? c2 : 0) * DI_;
  const float wmask2 = (c2 < 44) ? 1.f : 0.f;

  v8f acc0 = {}, acc1 = {}, acc2 = {};
#pragma unroll 2
  for (int k0 = 0; k0 < DI_; k0 += 4) {
    const v2f a  = *(const v2f*)(arow + k0 + half);
    const v2f b0 = *(const v2f*)(wrow0 + k0 + half);
    const v2f b1 = *(const v2f*)(wrow1 + k0 + half);
    v2f b2       = *(const v2f*)(wrow2 + k0 + half);
    b2.x *= wmask2; b2.y *= wmask2;
    acc0 = __builtin_amdgcn_wmma_f32_16x16x4_f32(false, a, false, b0, (short)0, acc0, false, false);
    acc1 = __builtin_amdgcn_wmma_f32_16x16x4_f32(false, a, false, b1, (short)0, acc1, false, false);
    acc2 = __builtin_amdgcn_wmma_f32_16x16x4_f32(false, a, false, b2, (short)0, acc2, false, false);
  }
  const int rofs = (lane < 16) ? 0 : 8;
  float* crow = xdbl + ((size_t)bk * L_ + tileM * 16) * NCH + lm;
#pragma unroll
  for (int r = 0; r < 8; ++r) {
    float* cr = crow + (size_t)(r + rofs) * NCH;
    cr[0]  = acc0[r];
    cr[16] = acc1[r];
    cr[32] = acc2[r];
  }
}

// ---------------------------------------------------------------------------
// Selective scan: one block per (b,k); thread d owns channel d, h[16] in regs.
// x_dbl chunks staged LDS-side with GLOBAL_LOAD_ASYNC_TO_LDS_B128 (ASYNCcnt),
// one 16B transfer per lane per chunk. Output scattered back to the original
// spatial position with global f32 atomics (6 directions summed).
// ---------------------------------------------------------------------------
__global__ __launch_bounds__(192)
void scan_kernel(const float* __restrict__ xsm, const float* __restrict__ xdbl,
                 const float* __restrict__ dtw_all, const float* __restrict__ dtb_all,
                 const float* __restrict__ Alogs, const float* __restrict__ Dsv,
                 float* __restrict__ ycomb) {
  __shared__ __align__(16) float sbuf[CHUNK * NCH];   // 768 floats = 3 KB
  const int bk = blockIdx.x;
  const int b = bk / 6, k = bk % 6;
  const int p = (k >= 3) ? (k - 3) : k;
  const int d = threadIdx.x;

  float An[DSN], hn[DSN];
#pragma unroll
  for (int n = 0; n < DSN; ++n) {
    An[n] = -expf(Alogs[(size_t)(k * DI_ + d) * DSN + n]);
    hn[n] = 0.f;
  }
  float dtw[DR_];
#pragma unroll
  for (int r = 0; r < DR_; ++r) dtw[r] = dtw_all[(size_t)(k * DI_ + d) * DR_ + r];
  const float dtb = dtb_all[k * DI_ + d];
  const float Dd  = Dsv[k * DI_ + d];

  const float* xsrc = xsm + (size_t)(b * 3 + p) * L_ * DI_;
  const float* xdk  = xdbl + (size_t)bk * L_ * NCH;
  float* yb = ycomb + (size_t)b * L_ * DI_;

  // per-lane LDS byte address for the async staging (generic ptr low 32 bits
  // are the LDS offset per the gfx1250 aperture mapping)
  const unsigned lds_addr = (unsigned)(uintptr_t)(&sbuf[d * 4]);

  for (int lc = 0; lc < L_; lc += CHUNK) {
    // async stage: chunk is 768 contiguous floats; lane d copies 16 bytes
    const float* gsrc = xdk + (size_t)lc * NCH + d * 4;
    asm volatile("global_load_async_to_lds_b128 %0, %1, off"
                 :: "v"(lds_addr), "v"(gsrc) : "memory");
    if (lc + CHUNK < L_) {
      const int nl = (k < 3) ? (lc + CHUNK) : (L_ - 1 - (lc + CHUNK));
      __builtin_prefetch(xsrc + (size_t)nl * DI_ + d, 0, 1);
    }
    asm volatile("s_wait_asynccnt 0x0" ::: "memory");
    __syncthreads();

    for (int ll = 0; ll < CHUNK; ++ll) {
      const int l    = lc + ll;
      const int lpos = (k < 3) ? l : (L_ - 1 - l);
      const float x_t = xsrc[(size_t)lpos * DI_ + d];
      const float* row = sbuf + ll * NCH;

      float s = dtb;
#pragma unroll
      for (int r = 0; r < DR_; ++r) s += row[r] * dtw[r];
      const float delta = (s > 20.f) ? s : log1pf(expf(s));   // softplus
      const float dx = delta * x_t;

      float y = 0.f;
#pragma unroll
      for (int n = 0; n < DSN; ++n) {
        hn[n] = expf(delta * An[n]) * hn[n] + dx * row[DR_ + n];
        y += hn[n] * row[DR_ + DSN + n];
      }
      const float outv = y + Dd * x_t;

      // map scan position back to spatial flat index (inverse permutation)
      const int a = lpos >> 8, b2 = (lpos >> 4) & 15, cc = lpos & 15;
      const int sp = (p == 0) ? lpos
                   : (p == 1) ? ((cc << 8) + (b2 << 4) + a)
                              : ((a  << 8) + (cc << 4) + b2);
      atomicAdd(yb + (size_t)sp * DI_ + d, outv);
    }
    __syncthreads();
  }
}

// ---------------------------------------------------------------------------
__global__ void zero_kernel(float* __restrict__ p, int n) {
  const int i = blockIdx.x * blockDim.x + threadIdx.x;
  if (i < n) p[i] = 0.f;
}

// LayerNorm over DI=192 per (b,l); LDS tree reduction (192 = 6 x wave32).
__global__ __launch_bounds__(192)
void ln_kernel(const float* __restrict__ y, const float* __restrict__ g,
               const float* __restrict__ bb, float* __restrict__ out) {
  __shared__ float red[192];
  __shared__ float smu, srstd;
  const int bl = blockIdx.x;
  const int d  = threadIdx.x;
  const float v = y[(size_t)bl * DI_ + d];

  red[d] = v;
  __syncthreads();
  for (int s = 96; s >= 3; s >>= 1) {       // 192->96->48->24->12->6->3
    if (d < s) red[d] += red[d + s];
    __syncthreads();
  }
  if (d == 0) smu = (red[0] + red[1] + red[2]) * (1.f / 192.f);
  __syncthreads();
  const float c = v - smu;

  red[d] = c * c;
  __syncthreads();
  for (int s = 96; s >= 3; s >>= 1) {
    if (d < s) red[d] += red[d + s];
    __syncthreads();
  }
  if (d == 0) srstd = rsqrtf((red[0] + red[1] + red[2]) * (1.f / 192.f) + 1e-5f);
  __syncthreads();

  out[(size_t)bl * DI_ + d] = c * srstd * g[d] + bb[d];
}

// ---------------------------------------------------------------------------
extern "C" void kernel_launch(void* const* d_in, const int* in_sizes, int n_in,
                              void* d_out, int out_size, void* d_ws, size_t ws_size,
                              hipStream_t stream) {
  const float* x    = (const float*)d_in[0];   // (B,16,16,16,192)
  const float* ipw  = (const float*)d_in[1];   // (192,192)
  const float* cw   = (const float*)d_in[2];   // (192,1,3,3,3)
  const float* cb   = (const float*)d_in[3];   // (192)
  const float* xpw  = (const float*)d_in[4];   // (6,44,192)
  const float* dtw  = (const float*)d_in[5];   // (6,192,12)
  const float* dtb  = (const float*)d_in[6];   // (6,192)
  const float* alog = (const float*)d_in[7];   // (1152,16)
  const float* Dsv  = (const float*)d_in[8];   // (1152)
  const float* lng  = (const float*)d_in[9];   // (192)
  const float* lnb  = (const float*)d_in[10];  // (192)
  const float* opw  = (const float*)d_in[11];  // (192,192)

  // Workspace layout (stream-ordered aliasing keeps usage ~35 MB):
  //   buf0: xsm   (B,3,L,192)                 18.9 MB
  //   buf1: xdbl  (B,6,L,48)  -> ynorm reuse   9.4 MB
  //   buf2: xi    (B,L,192)   -> ycomb reuse   6.3 MB
  char* ws = (char*)d_ws;
  float* xsm  = (float*)ws;
  float* xdbl = (float*)(ws + (size_t)B_ * 3 * L_ * DI_ * 4);
  float* xi   = (float*)(ws + (size_t)B_ * 3 * L_ * DI_ * 4
                            + (size_t)B_ * 6 * L_ * NCH * 4);
  float* ycmb = xi;    // xi dead after conv
  float* ynrm = xdbl;  // xdbl dead after scan

  const int M = B_ * L_;  // 8192 rows

  // 1) in_proj GEMM (WMMA, 16x64 strips): x -> xi
  gemm_xtw_kernel<<<dim3(M / 16, DM_ / 64), 32, 0, stream>>>(x, ipw, xi);
  // 2) depthwise conv + SiLU + scan-order scatter: xi -> xsm
  conv_silu_perm_kernel<<<M, DI_, 0, stream>>>(xi, cw, cb, xsm);
  // 3) zero the combine buffer (aliases xi, now dead)
  zero_kernel<<<(M * DI_ + 255) / 256, 256, 0, stream>>>(ycmb, M * DI_);
  // 4) x_dbl projection GEMM (WMMA, 16x48 strips), all 6 directions
  gemm_xdbl_kernel<<<dim3(L_ / 16, B_ * K_), 32, 0, stream>>>(xsm, xpw, xdbl);
  // 5) selective scan + inverse-permute accumulate (async LDS staging)
  scan_kernel<<<B_ * K_, DI_, 0, stream>>>(xsm, xdbl, dtw, dtb, alog, Dsv, ycmb);
  // 6) LayerNorm over channels
  ln_kernel<<<M, DI_, 0, stream>>>(ycmb, lng, lnb, ynrm);
  // 7) out_proj GEMM (WMMA, 16x64 strips) -> d_out
  gemm_xtw_kernel<<<dim3(M / 16, DM_ / 64), 32, 0, stream>>>(ynrm, opw, (float*)d_out);
}